// BatchHardTripletLoss_42176578847371
// MI455X (gfx1250) — compile-verified
//
#include <hip/hip_runtime.h>

// Problem constants (from reference)
#define N_ 4096
#define D_ 512
#define MARGIN_ 0.3f
#define AS_STRIDE 516   // 512 + 4 pad: per-lane A reads hit distinct LDS banks

typedef __attribute__((ext_vector_type(2))) float v2f;
typedef __attribute__((ext_vector_type(8))) float v8f;

// ---------------------------------------------------------------------------
// Kernel 1: L2-normalize each row (x / max(||x||, 1e-12)) into workspace.
// ---------------------------------------------------------------------------
__global__ __launch_bounds__(128)
void tl_normalize(const float* __restrict__ emb, float* __restrict__ en) {
    const int row  = blockIdx.x;
    const int t    = threadIdx.x;            // 0..127
    const int wave = t >> 5, lane = t & 31;

    const float4* rp = (const float4*)(emb + (size_t)row * D_);
    float4 v = rp[t];                        // 128 * 4 = 512 floats
    float ss = v.x * v.x + v.y * v.y + v.z * v.z + v.w * v.w;

    #pragma unroll
    for (int off = 16; off >= 1; off >>= 1)
        ss += __shfl_xor(ss, off, 32);

    __shared__ float wss[4];
    __shared__ float s_scale;
    if (lane == 0) wss[wave] = ss;
    __syncthreads();
    if (t == 0) {
        float tot = wss[0] + wss[1] + wss[2] + wss[3];
        s_scale = 1.0f / fmaxf(sqrtf(tot), 1e-12f);
    }
    __syncthreads();

    const float sc = s_scale;
    float4 o; o.x = v.x * sc; o.y = v.y * sc; o.z = v.z * sc; o.w = v.w * sc;
    ((float4*)(en + (size_t)row * D_))[t] = o;
}

// ---------------------------------------------------------------------------
// Kernel 2: fused Gram (fp32 WMMA 16x16x4) + masked hardest-pos/neg reduce.
// 256 blocks x 256 threads (8 waves). Block b owns rows [16b, 16b+16).
// Each wave processes TWO column tiles at once (shared A strip, two
// independent WMMA accumulator chains for ILP on the matrix pipe).
// dist = sqrt(clip(2-2g)) is monotone non-increasing in g, so we reduce on
// the Gram values directly (min over positives / max over negatives) and
// apply sqrt once per row at the end.
// ---------------------------------------------------------------------------
__global__ __launch_bounds__(256)
void tl_gram_reduce(const float* __restrict__ en, const int* __restrict__ lab,
                    float* __restrict__ rowAP, float* __restrict__ rowAN) {
    __shared__ float As[16 * AS_STRIDE];     // padded A strip (16 x 512)
    __shared__ int   lrow[16];
    __shared__ float apW[8][16];             // per-wave min-g over positives
    __shared__ float anW[8][16];             // per-wave max-g over negatives

    const int tid = threadIdx.x;
    const int rb  = blockIdx.x * 16;

    // Cooperative load of the 16x512 A strip (float4 granules).
    for (int idx = tid; idx < 16 * (D_ / 4); idx += 256) {
        const int r  = idx >> 7;             // /128
        const int c4 = idx & 127;
        float4 v = ((const float4*)(en + (size_t)(rb + r) * D_))[c4];
        *(float4*)&As[r * AS_STRIDE + c4 * 4] = v;
    }
    if (tid < 16) lrow[tid] = lab[rb + tid];
    __syncthreads();

    const int wave = tid >> 5;
    const int lane = tid & 31;
    const int ln   = lane & 15;              // A row / B col within tile
    const int hi   = lane >> 4;              // K sub-offset select

    float apg[8], ang[8];                    // min-g(pos), max-g(neg) per row
    #pragma unroll
    for (int r = 0; r < 8; ++r) { apg[r] = 1e30f; ang[r] = -1e30f; }

    const float* a_lds = &As[ln * AS_STRIDE + 2 * hi];

    for (int ct = wave; ct < N_ / 16; ct += 16) {
        const int cb0 = ct * 16;
        const int cb1 = (ct + 8) * 16;
        const int lc0 = lab[cb0 + ln];
        const int lc1 = lab[cb1 + ln];
        const float* bp0 = en + (size_t)(cb0 + ln) * D_ + 2 * hi;
        const float* bp1 = en + (size_t)(cb1 + ln) * D_ + 2 * hi;

        v8f c0 = {0.f, 0.f, 0.f, 0.f, 0.f, 0.f, 0.f, 0.f};
        v8f c1 = {0.f, 0.f, 0.f, 0.f, 0.f, 0.f, 0.f, 0.f};
        #pragma unroll 4
        for (int k = 0; k < D_; k += 4) {
            float2 ta = *(const float2*)(a_lds + k);  // A: M=ln, K=k+2hi..+1
            float2 t0 = *(const float2*)(bp0 + k);    // B0: N=ln
            float2 t1 = *(const float2*)(bp1 + k);    // B1: N=ln
            v2f a;  a[0]  = ta.x; a[1]  = ta.y;
            v2f b0; b0[0] = t0.x; b0[1] = t0.y;
            v2f b1; b1[0] = t1.x; b1[1] = t1.y;
            c0 = __builtin_amdgcn_wmma_f32_16x16x4_f32(
                     false, a, false, b0, (short)0, c0, false, false);
            c1 = __builtin_amdgcn_wmma_f32_16x16x4_f32(
                     false, a, false, b1, (short)0, c1, false, false);
        }

        // C tile: VGPR r holds (M=r, N=ln) in lanes 0-15, (M=r+8, N=ln) in 16-31.
        #pragma unroll
        for (int r = 0; r < 8; ++r) {
            const int m  = r + 8 * hi;
            const int gi = rb + m;
            const int rl = lrow[m];

            const float g0  = c0[r];
            const bool pos0 = (rl == lc0) && (gi != cb0 + ln);
            const bool neg0 = (rl != lc0);
            apg[r] = fminf(apg[r], pos0 ? g0 : 1e30f);
            ang[r] = fmaxf(ang[r], neg0 ? g0 : -1e30f);

            const float g1  = c1[r];
            const bool pos1 = (rl == lc1) && (gi != cb1 + ln);
            const bool neg1 = (rl != lc1);
            apg[r] = fminf(apg[r], pos1 ? g1 : 1e30f);
            ang[r] = fmaxf(ang[r], neg1 ? g1 : -1e30f);
        }
    }

    // Reduce over the 16 columns held by each half-wave (masks 8..1 stay in-half).
    #pragma unroll
    for (int r = 0; r < 8; ++r) {
        #pragma unroll
        for (int off = 8; off >= 1; off >>= 1) {
            apg[r] = fminf(apg[r], __shfl_xor(apg[r], off, 32));
            ang[r] = fmaxf(ang[r], __shfl_xor(ang[r], off, 32));
        }
    }
    if (ln == 0) {                                           // lanes 0 and 16
        #pragma unroll
        for (int r = 0; r < 8; ++r) {
            const int m = r + 8 * hi;
            apW[wave][m] = apg[r];
            anW[wave][m] = ang[r];
        }
    }
    __syncthreads();

    if (tid < 16) {                                          // cross-wave reduce
        float gp = 1e30f, gn = -1e30f;
        #pragma unroll
        for (int w = 0; w < 8; ++w) {
            gp = fminf(gp, apW[w][tid]);
            gn = fmaxf(gn, anW[w][tid]);
        }
        const bool has_pos = (gp < 1e29f);
        const bool has_neg = (gn > -1e29f);
        // Safe sqrt distance, applied once per row (monotone in g).
        const float sqp = fmaxf(2.0f - 2.0f * gp, 0.0f);
        const float sqn = fmaxf(2.0f - 2.0f * gn, 0.0f);
        const float dap = (sqp > 0.0f) ? sqrtf(sqp) : 0.0f;
        const float dan = (sqn > 0.0f) ? sqrtf(sqn) : 0.0f;
        rowAP[rb + tid] = has_pos ? dap : -1e30f;            // sentinel if no pos
        rowAN[rb + tid] = has_neg ? dan : 1e30f;             // sentinel if no neg
    }
}

// ---------------------------------------------------------------------------
// Kernel 3: deterministic final reduction -> masked mean of relu(ap-an+margin).
// ---------------------------------------------------------------------------
__global__ __launch_bounds__(256)
void tl_finalize(const float* __restrict__ rowAP, const float* __restrict__ rowAN,
                 float* __restrict__ out) {
    __shared__ float ssum[256];
    __shared__ float scnt[256];
    const int t = threadIdx.x;
    float s = 0.0f, c = 0.0f;
    for (int i = t; i < N_; i += 256) {
        const float a_ = rowAP[i], b_ = rowAN[i];
        if (a_ > -1e29f && b_ < 1e29f) {                     // row has pos & neg
            s += fmaxf(a_ - b_ + MARGIN_, 0.0f);
            c += 1.0f;
        }
    }
    ssum[t] = s; scnt[t] = c;
    __syncthreads();
    #pragma unroll
    for (int off = 128; off >= 1; off >>= 1) {
        if (t < off) { ssum[t] += ssum[t + off]; scnt[t] += scnt[t + off]; }
        __syncthreads();
    }
    if (t == 0)
        out[0] = (scnt[0] > 0.0f) ? ssum[0] / fmaxf(scnt[0], 1.0f) : 0.0f;
}

// ---------------------------------------------------------------------------
extern "C" void kernel_launch(void* const* d_in, const int* in_sizes, int n_in,
                              void* d_out, int out_size, void* d_ws, size_t ws_size,
                              hipStream_t stream) {
    (void)in_sizes; (void)n_in; (void)out_size; (void)ws_size;

    const float* emb = (const float*)d_in[0];   // [N, D] f32
    const int*   lab = (const int*)d_in[1];     // [N] labels
    float*       out = (float*)d_out;           // scalar f32

    float* en    = (float*)d_ws;                // [N, D] normalized
    float* rowAP = en + (size_t)N_ * D_;        // [N]
    float* rowAN = rowAP + N_;                  // [N]

    tl_normalize  <<<N_,      128, 0, stream>>>(emb, en);
    tl_gram_reduce<<<N_ / 16, 256, 0, stream>>>(en, lab, rowAP, rowAN);
    tl_finalize   <<<1,       256, 0, stream>>>(rowAP, rowAN, out);
}